// FETANetwork_14542759264829
// MI455X (gfx1250) — compile-verified
//
#include <hip/hip_runtime.h>

typedef __attribute__((ext_vector_type(2))) float v2f;
typedef __attribute__((ext_vector_type(8))) float v8f;

#define SELU_SCALE 1.0507009873554805f
#define SELU_ALPHA 1.6732632423543772f

// Branchless: selu(x) = scale*(max(x,0) + alpha*(exp(min(x,0))-1))
//   x>0: exp(0)-1 = 0 -> scale*x ;  x<=0: scale*alpha*(exp(x)-1)
__device__ __forceinline__ float selu_f(float x) {
    const float e = __expf(fminf(x, 0.0f));
    return SELU_SCALE * fmaf(SELU_ALPHA, e - 1.0f, fmaxf(x, 0.0f));
}

struct alignas(16) SharedBlk {
    float x[64 * 128];     // 32 KB: x[b] tile
    float WT[24 * 128];    // 12 KB: transposed cols: 0-7 Wz0, 8-15 Wa, 16-23 Wb
    float Wz1[64];
    float Wf1[64];
    float bz0[8], bz1[8], bf0[8], bf1[8], wsz[8], wl[8], wr[8];
    float bszv, bsfv;
    float hz[64 * 8];      // selu(x@Wz0+bz0)
    float hi[64 * 8];      // x@Wa (no bias)
    float hj[64 * 8];      // x@Wb (no bias)
    float zeroth[64];
    float rowsum[64];      // sum_j  f[i,j].wl  (j != i)
    float colsum[64];      // sum_i  f[i,j].wr  (i != j), indexed by j
};

__global__ __launch_bounds__(256) void feta_kernel(
    const float* __restrict__ x,   const float* __restrict__ Wz0,
    const float* __restrict__ bz0, const float* __restrict__ Wz1,
    const float* __restrict__ bz1, const float* __restrict__ wsz,
    const float* __restrict__ bsz, const float* __restrict__ Wf0,
    const float* __restrict__ bf0, const float* __restrict__ Wf1,
    const float* __restrict__ bf1, const float* __restrict__ wsf,
    const float* __restrict__ bsf, float* __restrict__ out)
{
    __shared__ SharedBlk s;
    const int t = threadIdx.x;
    const int b = blockIdx.x;

    // ---- Stage: x tile (coalesced b128), transposed projection weights, smalls
    {
        const float4* xg4 = (const float4*)(x + (size_t)b * (64 * 128));
        float4* xs4 = (float4*)s.x;
        for (int v = t; v < 2048; v += 256) xs4[v] = xg4[v];
    }
    for (int idx = t; idx < 24 * 128; idx += 256) {
        const int c = idx >> 7, k = idx & 127;
        float w;
        if (c < 8)       w = Wz0[k * 8 + c];
        else if (c < 16) w = Wf0[k * 8 + (c - 8)];
        else             w = Wf0[(128 + k) * 8 + (c - 16)];
        s.WT[idx] = w;
    }
    if (t < 64) { s.Wz1[t] = Wz1[t]; s.Wf1[t] = Wf1[t]; }
    if (t < 8) {
        s.bz0[t] = bz0[t]; s.bz1[t] = bz1[t]; s.bf0[t] = bf0[t]; s.bf1[t] = bf1[t];
        s.wsz[t] = wsz[t]; s.wl[t]  = wsf[t]; s.wr[t]  = wsf[8 + t];
    }
    if (t == 0) { s.bszv = bsz[0]; s.bsfv = bsf[0]; }
    __syncthreads();

    // ---- Phase 1: hz = selu(x@Wz0+bz0), hi = x@Wa, hj = x@Wb  (24 cols x 64 rows)
    for (int task = t; task < 64 * 24; task += 256) {
        const int row = task / 24, col = task % 24;
        const float4* x4 = (const float4*)&s.x[row * 128];
        const float4* w4 = (const float4*)&s.WT[col * 128];
        float acc = 0.0f;
#pragma unroll
        for (int kk = 0; kk < 32; ++kk) {
            const float4 xv = x4[kk], wv = w4[kk];
            acc = fmaf(xv.x, wv.x, acc);
            acc = fmaf(xv.y, wv.y, acc);
            acc = fmaf(xv.z, wv.z, acc);
            acc = fmaf(xv.w, wv.w, acc);
        }
        if (col < 8)       s.hz[row * 8 + col]        = selu_f(acc + s.bz0[col]);
        else if (col < 16) s.hi[row * 8 + (col - 8)]  = acc;
        else               s.hj[row * 8 + (col - 16)] = acc;
    }
    __syncthreads();

    // ---- Phase 2: zeroth path; zero pair accumulators
    if (t < 64) {
        float zacc = s.bszv;
#pragma unroll
        for (int f = 0; f < 8; ++f) {
            float a = s.bz1[f];
#pragma unroll
            for (int k = 0; k < 8; ++k) a = fmaf(s.hz[t * 8 + k], s.Wz1[k * 8 + f], a);
            zacc = fmaf(selu_f(a), s.wsz[f], zacc);
        }
        s.zeroth[t] = selu_f(zacc);
        s.rowsum[t] = 0.0f;
        s.colsum[t] = 0.0f;
    }
    __syncthreads();

    // ---- Phase 3: pairwise (B,N,N,8)@Wf1 via V_WMMA_F32_16X16X4_F32
    // A (16x4/k-step): A[m][k] = Wf1[k][m%8]  (Wf1^T, rows duplicated into M=16)
    // B (4x16/k-step): B[k][n] = g[pair n][k], g = selu(hi[i]+hj[j]+bf0)
    // D (16x16): v[feat, pair], feat = accumulator VGPR index, identical in both
    // lane halves -> wl/wr dots are pure per-lane register loops.
    {
        const int lane  = t & 31;
        const int wave  = t >> 5;                // 8 waves
        const int n16   = lane & 15;
        const int kb    = (lane >= 16) ? 2 : 0;  // K-offset for upper lane half
        const int mfeat = n16 & 7;

        v2f a0, a1;                               // constant A fragments (k-steps 0,1)
        a0.x = s.Wf1[(kb + 0) * 8 + mfeat];
        a0.y = s.Wf1[(kb + 1) * 8 + mfeat];
        a1.x = s.Wf1[(kb + 4) * 8 + mfeat];
        a1.y = s.Wf1[(kb + 5) * 8 + mfeat];

        // epilogue constants -> registers (no per-tile uniform DS loads)
        float bf1v[8], wlv[8], wrv[8];
#pragma unroll
        for (int r = 0; r < 8; ++r) {
            bf1v[r] = s.bf1[r]; wlv[r] = s.wl[r]; wrv[r] = s.wr[r];
        }

        // hj (+bf0 folded) preloaded once: each lane owns 4 fixed j columns
        float hjv[4][4];
#pragma unroll
        for (int jb = 0; jb < 4; ++jb) {
            const int j = jb * 16 + n16;
            hjv[jb][0] = s.hj[j * 8 + kb + 0] + s.bf0[kb + 0];
            hjv[jb][1] = s.hj[j * 8 + kb + 1] + s.bf0[kb + 1];
            hjv[jb][2] = s.hj[j * 8 + kb + 4] + s.bf0[kb + 4];
            hjv[jb][3] = s.hj[j * 8 + kb + 5] + s.bf0[kb + 5];
        }

        float csum[4] = {0.0f, 0.0f, 0.0f, 0.0f};  // per-lane colsum accumulators

        for (int ii = 0; ii < 8; ++ii) {
            const int i = wave * 8 + ii;           // wave-exclusive i range
            const float hiv0 = s.hi[i * 8 + kb + 0];
            const float hiv1 = s.hi[i * 8 + kb + 1];
            const float hiv2 = s.hi[i * 8 + kb + 4];
            const float hiv3 = s.hi[i * 8 + kb + 5];

            float rsum = 0.0f;
#pragma unroll
            for (int jb = 0; jb < 4; ++jb) {
                v2f b0, b1;
                b0.x = selu_f(hiv0 + hjv[jb][0]);
                b0.y = selu_f(hiv1 + hjv[jb][1]);
                b1.x = selu_f(hiv2 + hjv[jb][2]);
                b1.y = selu_f(hiv3 + hjv[jb][3]);

                v8f acc = {};
                acc = __builtin_amdgcn_wmma_f32_16x16x4_f32(false, a0, false, b0,
                                                            (short)0, acc, false, false);
                acc = __builtin_amdgcn_wmma_f32_16x16x4_f32(false, a1, false, b1,
                                                            (short)0, acc, false, false);

                float sl = 0.0f, sr = 0.0f;
#pragma unroll
                for (int r = 0; r < 8; ++r) {
                    const float v = selu_f(acc[r] + bf1v[r]);
                    sl = fmaf(v, wlv[r], sl);
                    sr = fmaf(v, wrv[r], sr);
                }
                const bool diag = (jb * 16 + n16) == i;   // mask j == i
                rsum += diag ? 0.0f : sl;
                csum[jb] += diag ? 0.0f : sr;
            }
            // one 16-lane reduction per i (4 tiles folded in already)
#pragma unroll
            for (int off = 1; off < 16; off <<= 1)
                rsum += __shfl_xor(rsum, off, 16);
            if (lane == 0) atomicAdd(&s.rowsum[i], rsum);
        }
        // flush colsum: 4 atomics per wave, lanes 0-15 only (halves are duplicates)
        if (lane < 16) {
#pragma unroll
            for (int jb = 0; jb < 4; ++jb)
                atomicAdd(&s.colsum[jb * 16 + n16], csum[jb]);
        }
    }
    __syncthreads();

    // ---- Phase 4: pair_mean + zeroth
    if (t < 64) {
        out[(size_t)b * 64 + t] =
            (s.rowsum[t] + s.colsum[t]) * (1.0f / 63.0f) + s.bsfv + s.zeroth[t];
    }
}

extern "C" void kernel_launch(void* const* d_in, const int* in_sizes, int n_in,
                              void* d_out, int out_size, void* d_ws, size_t ws_size,
                              hipStream_t stream) {
    const float* x   = (const float*)d_in[0];
    const float* Wz0 = (const float*)d_in[1];
    const float* bz0 = (const float*)d_in[2];
    const float* Wz1 = (const float*)d_in[3];
    const float* bz1 = (const float*)d_in[4];
    const float* wsz = (const float*)d_in[5];
    const float* bsz = (const float*)d_in[6];
    const float* Wf0 = (const float*)d_in[7];
    const float* bf0 = (const float*)d_in[8];
    const float* Wf1 = (const float*)d_in[9];
    const float* bf1 = (const float*)d_in[10];
    const float* wsf = (const float*)d_in[11];
    const float* bsf = (const float*)d_in[12];
    float* out = (float*)d_out;

    feta_kernel<<<1024, 256, 0, stream>>>(x, Wz0, bz0, Wz1, bz1, wsz, bsz,
                                          Wf0, bf0, Wf1, bf1, wsf, bsf, out);
}